// BiLSTM_CRF_54769422958953
// MI455X (gfx1250) — compile-verified
//
#include <hip/hip_runtime.h>

#define NUM_TAGS 5
#define T_LEN 512
#define START_TAG 3
#define STOP_TAG 4
#define TT 16                 // timesteps per LDS tile
#define RENORM 4              // power-of-two renormalization every RENORM steps
#define SEQ_PER_WAVE 16
#define WAVES_PER_BLOCK 4
#define SEQ_PER_BLOCK 64
#define BLOCK_THREADS 128
#define ROWF 84               // padded LDS row (floats) per sequence: 16*5 + 4 pad
#define LOG2E 1.4426950408889634f
#define LN2   0.6931471805599453f
#define TINY  1e-38f

typedef __attribute__((ext_vector_type(2))) float v2f;
typedef __attribute__((ext_vector_type(8))) float v8f;

__global__ __launch_bounds__(BLOCK_THREADS)
void crf_forward_wmma(const float* __restrict__ feats,
                      const float* __restrict__ trans,
                      float* __restrict__ out) {
  __shared__ float sh[SEQ_PER_BLOCK * ROWF];

  const int tid     = threadIdx.x;
  const int lane    = tid & 31;
  const int n16     = lane & 15;
  const int half    = lane >> 4;
  const int waveId  = tid >> 5;
  const int seqLocal = waveId * SEQ_PER_WAVE + n16;
  const long b0 = (long)blockIdx.x * SEQ_PER_BLOCK;
  const long b  = b0 + seqLocal;

  // ---- A operands: E = exp(transitions) in 16x4 f32 WMMA A layout ----
  // Rows 0-4 = E rows {0,1,2,3,4}. Rows 8-12 = E rows {2,3,0,1,4} (pair-swap
  // permutation), so the upper 16 lanes' registers carry tags {2,3,0,1,4}.
  // With that, bb1 = {v0, v1} feeds K0/K1 (lower: tags 0/1) and K2/K3
  // (upper: tags 2/3) with zero per-step selects.
  // A1 covers prev(K)=0..3, A2 has prev=4 in its K=0 slot (other K cols zero).
  v2f a1, a2;
  {
    const int  kk = n16 & 7;             // row within each 8-row copy
    const bool rv = kk < NUM_TAGS;       // rows 0-4 and 8-12 hold E
    int r = (n16 < 8) ? kk : ((kk < 4) ? (kk ^ 2) : kk);  // perm {2,3,0,1,4}
    r = rv ? r : 0;
    const int c0 = half * 2;             // K for VGPR0: 0 (lanes 0-15) / 2 (16-31)
    const int c1 = half * 2 + 1;         // K for VGPR1: 1 / 3
    float e0 = __builtin_amdgcn_exp2f(trans[r * NUM_TAGS + c0] * LOG2E);
    float e1 = __builtin_amdgcn_exp2f(trans[r * NUM_TAGS + c1] * LOG2E);
    float e4 = __builtin_amdgcn_exp2f(trans[r * NUM_TAGS + 4]  * LOG2E);
    a1.x = rv ? e0 : 0.0f;
    a1.y = rv ? e1 : 0.0f;
    a2.x = (rv && half == 0) ? e4 : 0.0f;  // K=0 slot only
    a2.y = 0.0f;
  }

  // STOP row of exp(transitions) for terminal reduction (valid on lower lanes).
  const float es0 = __builtin_amdgcn_exp2f(trans[STOP_TAG * NUM_TAGS + 0] * LOG2E);
  const float es1 = __builtin_amdgcn_exp2f(trans[STOP_TAG * NUM_TAGS + 1] * LOG2E);
  const float es2 = __builtin_amdgcn_exp2f(trans[STOP_TAG * NUM_TAGS + 2] * LOG2E);
  const float es3 = __builtin_amdgcn_exp2f(trans[STOP_TAG * NUM_TAGS + 3] * LOG2E);
  const float es4 = __builtin_amdgcn_exp2f(trans[STOP_TAG * NUM_TAGS + 4] * LOG2E);

  // Linear-domain state: true_fv[k] = log2(v_tag(k)) + off (off = sum of
  // power-of-two renorm exponents). init fv: START=0 -> v[START]=1, rest 0.
  // Upper lanes hold the permuted tag order {2,3,0,1,4}.
  float v0 = 0.0f;
  float v1 = half ? 1.0f : 0.0f;   // upper slot1 = tag 3 = START
  float v2 = 0.0f;
  float v3 = half ? 0.0f : 1.0f;   // lower slot3 = tag 3 = START
  float v4 = 0.0f;
  float off = 0.0f;

  const float* rowp  = &sh[seqLocal * ROWF];
  const float* baseA = rowp + (half ? 2 : 0);  // slots 0,1: tags {0,1}/{2,3}
  const float* baseB = rowp + (half ? 0 : 2);  // slots 2,3: tags {2,3}/{0,1}

  for (int t0 = 0; t0 < T_LEN; t0 += TT) {
    __syncthreads();
    // Cooperative, coalesced tile load: 64 seq x 80 floats = 1280 float4s,
    // 10 per thread. Feats pre-scaled by LOG2E (off the serial chain).
    #pragma unroll
    for (int it = 0; it < 10; ++it) {
      int i = tid + it * BLOCK_THREADS;
      int s = i / 20;             // local sequence
      int p = i - s * 20;         // float4 piece within 320B chunk
      const float* gp = feats + (b0 + s) * (long)(T_LEN * NUM_TAGS)
                              + t0 * NUM_TAGS + (p << 2);
      float4 v = *reinterpret_cast<const float4*>(gp);
      v.x *= LOG2E; v.y *= LOG2E; v.z *= LOG2E; v.w *= LOG2E;
      *reinterpret_cast<float4*>(&sh[s * ROWF + (p << 2)]) = v;
    }
    __syncthreads();

    // Hint-prefetch the next tile (global_prefetch_b8 pulls full cachelines).
    if (t0 + TT < T_LEN) {
      int s = tid & 63, part = tid >> 6;
      const float* np = feats + (b0 + s) * (long)(T_LEN * NUM_TAGS)
                              + (t0 + TT) * NUM_TAGS + part * 40;
      __builtin_prefetch(np, 0, 1);
    }

    #pragma unroll
    for (int blk = 0; blk < TT / RENORM; ++blk) {
      // Emission weights for the whole 4-step block, hoisted: the LDS loads
      // issue a full block ahead of use and the exps batch here, so the
      // recurrence chain below touches neither DS nor transcendentals.
      float w[RENORM][NUM_TAGS];
      #pragma unroll
      for (int u = 0; u < RENORM; ++u) {
        const int t5 = (blk * RENORM + u) * NUM_TAGS;
        w[u][0] = __builtin_amdgcn_exp2f(baseA[t5 + 0]);
        w[u][1] = __builtin_amdgcn_exp2f(baseA[t5 + 1]);
        w[u][2] = __builtin_amdgcn_exp2f(baseB[t5 + 0]);
        w[u][3] = __builtin_amdgcn_exp2f(baseB[t5 + 1]);
        w[u][4] = __builtin_amdgcn_exp2f(rowp[t5 + 4]);
      }

      // Exact power-of-two renormalization: integer exponent extract, no
      // transcendental, no rounding. Per-step growth <= 2^19 -> 4 steps safe.
      {
        float mx = fmaxf(fmaxf(fmaxf(v0, v1), fmaxf(v2, v3)), v4);
        int e = ((__float_as_int(mx) >> 23) & 0xff) - 127;
        float sc = __int_as_float((127 - e) << 23);   // 2^-e
        off += (float)e;
        v0 *= sc; v1 *= sc; v2 *= sc; v3 *= sc; v4 *= sc;
      }

      #pragma unroll
      for (int u = 0; u < RENORM; ++u) {
        v2f bb1, bb2;
        bb1.x = v0;        // K0 (lower: tag0) / K2 (upper: tag2)
        bb1.y = v1;        // K1 (lower: tag1) / K3 (upper: tag3)
        bb2.x = v4;        // K0 slot of A2 (prev=4); upper hits zero K=2 col
        bb2.y = 0.0f;

        v8f c = {};
        v8f d = __builtin_amdgcn_wmma_f32_16x16x4_f32(
                    false, a1, false, bb1, (short)0, c, false, false);
        d = __builtin_amdgcn_wmma_f32_16x16x4_f32(
                    false, a2, false, bb2, (short)0, d, false, false);

        v0 = d[0] * w[u][0];
        v1 = d[1] * w[u][1];
        v2 = d[2] * w[u][2];
        v3 = d[3] * w[u][3];
        v4 = d[4] * w[u][4];
      }
    }
  }

  // Terminal: alpha = logsumexp(fv + trans[STOP,:]) = ln2*(log2(sum es*v)+off).
  // Values are at most 4 steps past a renorm -> sum < 2^90, no overflow.
  float ssum = es0 * v0 + es1 * v1 + es2 * v2 + es3 * v3 + es4 * v4;
  float alpha = (__builtin_amdgcn_logf(ssum + TINY) + off) * LN2;

  if (half == 0) out[b] = alpha;
}

extern "C" void kernel_launch(void* const* d_in, const int* in_sizes, int n_in,
                              void* d_out, int out_size, void* d_ws, size_t ws_size,
                              hipStream_t stream) {
  const float* feats = (const float*)d_in[0];   // [B, T, K] f32
  const float* trans = (const float*)d_in[1];   // [K, K] f32
  float* out = (float*)d_out;                   // [B] f32
  (void)in_sizes; (void)n_in; (void)d_ws; (void)ws_size;

  int Btot = out_size;                          // 8192
  int blocks = Btot / SEQ_PER_BLOCK;            // 128 blocks x 128 threads
  crf_forward_wmma<<<blocks, BLOCK_THREADS, 0, stream>>>(feats, trans, out);
}